// Graph_Representation_Learning_Fedstar_68436008894717
// MI455X (gfx1250) — compile-verified
//
#include <hip/hip_runtime.h>

typedef __attribute__((ext_vector_type(2))) float v2f;
typedef __attribute__((ext_vector_type(8))) float v8f;

#define DIMC 128

// ---------------- elementwise / utility kernels ----------------

__global__ void fill_k(float* __restrict__ p, long n, float v) {
  long t = (long)blockIdx.x * blockDim.x + threadIdx.x;
  if (t < n) p[t] = v;
}

__global__ void degree_k(const int* __restrict__ dst, float* __restrict__ deg, long E) {
  long t = (long)blockIdx.x * blockDim.x + threadIdx.x;
  if (t < E) atomicAdd(&deg[dst[t]], 1.0f);
}

__global__ void dinv_k(float* __restrict__ d, long N) {
  long t = (long)blockIdx.x * blockDim.x + threadIdx.x;
  if (t < N) d[t] = rsqrtf(1.0f + d[t]);
}

// hc[i,0:128]=a[i], hc[i,128:256]=b[i]; optionally duplicate into agg (scatter accumulator init)
__global__ void concat2_k(const float* __restrict__ a, const float* __restrict__ b,
                          float* __restrict__ hc, float* __restrict__ agg, long n) {
  long t = (long)blockIdx.x * blockDim.x + threadIdx.x;
  if (t >= n) return;
  long i = t >> 7; int c = (int)(t & 127);
  float va = a[t], vb = b[t];
  size_t base = (size_t)i * 256 + c;
  hc[base] = va; hc[base + 128] = vb;
  if (agg) { agg[base] = va; agg[base + 128] = vb; }
}

// agg[dst] += hc[src]  (256 channels, 4 per thread)
__global__ void gin_scatter_k(const int* __restrict__ src, const int* __restrict__ dst,
                              const float* __restrict__ hc, float* __restrict__ agg, long E) {
  long t = (long)blockIdx.x * blockDim.x + threadIdx.x;
  long e = t >> 6;
  if (e >= E) return;
  int q = (int)(t & 63) * 4;
  int s = src[e], d = dst[e];
  const float4 v = *(const float4*)(hc + (size_t)s * 256 + q);
  float* p = agg + (size_t)d * 256 + q;
  atomicAdd(p + 0, v.x); atomicAdd(p + 1, v.y);
  atomicAdd(p + 2, v.z); atomicAdd(p + 3, v.w);
}

// BN statistics: per-channel sum and sum of squares via LDS atomics -> global atomics
__global__ void bn_stats_k(const float* __restrict__ h, float* __restrict__ stats, long n) {
  __shared__ float ls[256];
  if (threadIdx.x < 256) ls[threadIdx.x] = 0.0f;
  __syncthreads();
  long stride = (long)gridDim.x * blockDim.x;
  for (long t = (long)blockIdx.x * blockDim.x + threadIdx.x; t < n; t += stride) {
    float v = h[t];
    int c = (int)(t & 127);
    atomicAdd(&ls[c], v);
    atomicAdd(&ls[128 + c], v * v);
  }
  __syncthreads();
  if (threadIdx.x < 256) atomicAdd(&stats[threadIdx.x], ls[threadIdx.x]);
}

__global__ void bn_apply_k(const float* __restrict__ h, const float* __restrict__ stats,
                           const float* __restrict__ gamma, const float* __restrict__ beta,
                           float* __restrict__ out, long n, float invN) {
  long t = (long)blockIdx.x * blockDim.x + threadIdx.x;
  if (t >= n) return;
  int c = (int)(t & 127);
  float mu  = stats[c] * invN;
  float var = stats[128 + c] * invN - mu * mu;
  out[t] = (h[t] - mu) * rsqrtf(var + 1e-4f) * gamma[c] + beta[c];
}

// shn = sw * dinv^2 + bias (GCN self-loop term)
__global__ void gcn_base_k(const float* __restrict__ sw, const float* __restrict__ dinv,
                           const float* __restrict__ b, float* __restrict__ shn, long n) {
  long t = (long)blockIdx.x * blockDim.x + threadIdx.x;
  if (t >= n) return;
  long i = t >> 7; int c = (int)(t & 127);
  float di = dinv[i];
  shn[t] = sw[t] * di * di + b[c];
}

// shn[dst] += sw[src] * dinv[src]*dinv[dst]  (128 channels, 4 per thread)
__global__ void gcn_scatter_k(const int* __restrict__ src, const int* __restrict__ dst,
                              const float* __restrict__ sw, const float* __restrict__ dinv,
                              float* __restrict__ shn, long E) {
  long t = (long)blockIdx.x * blockDim.x + threadIdx.x;
  long e = t >> 5;
  if (e >= E) return;
  int q = (int)(t & 31) * 4;
  int s = src[e], d = dst[e];
  float coef = dinv[s] * dinv[d];
  const float4 v = *(const float4*)(sw + (size_t)s * 128 + q);
  float* p = shn + (size_t)d * 128 + q;
  atomicAdd(p + 0, v.x * coef); atomicAdd(p + 1, v.y * coef);
  atomicAdd(p + 2, v.z * coef); atomicAdd(p + 3, v.w * coef);
}

__global__ void tanh_k(const float* __restrict__ in, float* __restrict__ out, long n) {
  long t = (long)blockIdx.x * blockDim.x + threadIdx.x;
  if (t < n) out[t] = tanhf(in[t]);
}

// g[batch[i]] += y[i]
__global__ void pool_k(const float* __restrict__ y, const int* __restrict__ batch,
                       float* __restrict__ g, long n) {
  long t = (long)blockIdx.x * blockDim.x + threadIdx.x;
  if (t >= n) return;
  long i = t >> 7; int c = (int)(t & 127);
  atomicAdd(&g[(size_t)batch[i] * 128 + c], y[t]);
}

// ---------------- fp32 WMMA GEMM: out[M x 128] = act(A[M x K] @ W[K x 128] + bias + res) ----
// Block = 256 threads (8 waves). Each wave owns a 16-row x 128-col output strip
// (8 accumulator tiles of V_WMMA_F32_16X16X4_F32). K must be a multiple of 4.
__global__ __launch_bounds__(256) void wmma_gemm_k(
    const float* __restrict__ A, const float* __restrict__ W,
    const float* __restrict__ bias, const float* __restrict__ res,
    float* __restrict__ out, int M, int K, int act)
{
  const int lane = threadIdx.x & 31;
  const int wave = threadIdx.x >> 5;
  const int l16  = lane & 15;
  const int half = lane >> 4;                  // 0: lanes 0-15, 1: lanes 16-31
  const long rowBase = ((long)blockIdx.x * 8 + wave) * 16;

  v8f acc[8];
#pragma unroll
  for (int t = 0; t < 8; ++t)
#pragma unroll
    for (int j = 0; j < 8; ++j) acc[t][j] = 0.0f;

  // A fragment rows (clamped so out-of-range waves still execute with EXEC all-1s)
  long ar = rowBase + l16;
  if (ar > (long)M - 1) ar = (long)M - 1;
  const float* Arow = A + (size_t)ar * K;

  for (int kk = 0; kk < K; kk += 4) {
    // A 16x4 f32 fragment: VGPR0 = A[m][kk+2*half], VGPR1 = A[m][kk+2*half+1]
    v2f a;
    a.x = Arow[kk + 2 * half];
    a.y = Arow[kk + 2 * half + 1];
    const float* Wk = W + (size_t)(kk + 2 * half) * DIMC;
#pragma unroll
    for (int t = 0; t < 8; ++t) {
      const int col = t * 16 + l16;
      // B 4x16 f32 fragment (symmetric layout): rows kk+2*half, kk+2*half+1
      v2f b;
      b.x = Wk[col];
      b.y = Wk[DIMC + col];
      acc[t] = __builtin_amdgcn_wmma_f32_16x16x4_f32(
          /*neg_a=*/false, a, /*neg_b=*/false, b,
          /*c_mod=*/(short)0, acc[t], /*reuse_a=*/false, /*reuse_b=*/false);
    }
  }

  // Epilogue: D VGPR r -> row rowBase + r + 8*half, col = tile*16 + l16
#pragma unroll
  for (int t = 0; t < 8; ++t) {
    const int col = t * 16 + l16;
#pragma unroll
    for (int r = 0; r < 8; ++r) {
      long row = rowBase + r + half * 8;
      if (row < M) {
        float v = acc[t][r];
        if (bias) v += bias[col];
        if (res)  v += res[(size_t)row * DIMC + col];
        if (act)  v = (v >= 0.0f) ? v : 0.01f * v;   // LeakyReLU(0.01)
        out[(size_t)row * DIMC + col] = v;
      }
    }
  }
}

// ---------------- host orchestration ----------------

extern "C" void kernel_launch(void* const* d_in, const int* in_sizes, int n_in,
                              void* d_out, int out_size, void* d_ws, size_t ws_size,
                              hipStream_t stream) {
  (void)in_sizes; (void)n_in; (void)out_size; (void)ws_size;
  const long N = 100000, E = 800000, G = 500;

  const float* x     = (const float*)d_in[0];
  const float* s     = (const float*)d_in[1];
  const int*   ei    = (const int*)d_in[2];      // [2, E]: row0=src, row1=dst
  const int*   batch = (const int*)d_in[3];
  const float* W_pre = (const float*)d_in[4];
  const float* b_pre = (const float*)d_in[5];
  const float* W_s   = (const float*)d_in[6];
  const float* b_s   = (const float*)d_in[7];
  const float* ginW1 = (const float*)d_in[8];    // [3,256,128]
  const float* ginW2 = (const float*)d_in[9];    // [3,128,128]
  const float* bng   = (const float*)d_in[10];
  const float* bnb   = (const float*)d_in[11];
  const float* gcnW  = (const float*)d_in[12];   // [3,128,128]
  const float* gcnb  = (const float*)d_in[13];
  const float* W_hp  = (const float*)d_in[14];   // [256,128]
  const float* b_hp  = (const float*)d_in[15];
  const float* ffW1  = (const float*)d_in[16];
  const float* ffW2  = (const float*)d_in[17];
  const float* ffW3  = (const float*)d_in[18];
  const float* ffWs  = (const float*)d_in[19];
  float* out = (float*)d_out;

  // workspace layout (floats)
  float* ws   = (float*)d_ws;
  float* dinv = ws;                         // N
  float* h    = dinv + N;                   // N*128
  float* sh   = h + N * 128;                // N*128
  float* hloc = sh + N * 128;               // N*128
  float* hc   = hloc + N * 128;             // N*256
  float* agg  = hc + (size_t)N * 256;       // N*256
  float* stats= agg + (size_t)N * 256;      // 256
  float* g    = stats + 256;                // G*128
  float* gt1  = g + G * 128;                // G*128
  float* gt2  = gt1 + G * 128;              // G*128
  // reuse
  float* hidden = hc;                       // N*128 (hc consumed before write)
  float* sw     = hc + (size_t)N * 128;     // N*128
  float* shn    = agg;                      // N*128 (agg consumed before write)
  float* y      = agg;                      // N*128

  const int B = 256;
  auto blk = [](long n) { return (unsigned)((n + 255) / 256); };
  const unsigned gN = (unsigned)((N + 127) / 128);
  const unsigned gG = (unsigned)((G + 127) / 128);

  const int* src = ei;
  const int* dst = ei + E;

  // degree normalization: dinv = rsqrt(1 + in_degree)
  fill_k<<<blk(N), B, 0, stream>>>(dinv, N, 0.0f);
  degree_k<<<blk(E), B, 0, stream>>>(dst, dinv, E);
  dinv_k<<<blk(N), B, 0, stream>>>(dinv, N);

  // embeddings
  wmma_gemm_k<<<gN, B, 0, stream>>>(x, W_pre, b_pre, nullptr, h,  (int)N, 64, 0);
  wmma_gemm_k<<<gN, B, 0, stream>>>(s, W_s,   b_s,   nullptr, sh, (int)N, 16, 0);

  for (int i = 0; i < 3; ++i) {
    // GIN: agg = hc + scatter_add(hc[src] -> dst)
    concat2_k<<<blk(N * 128), B, 0, stream>>>(h, sh, hc, agg, N * 128);
    gin_scatter_k<<<blk(E * 64), B, 0, stream>>>(src, dst, hc, agg, E);
    // h = leaky(agg @ W1) @ W2
    wmma_gemm_k<<<gN, B, 0, stream>>>(agg, ginW1 + (size_t)i * 256 * 128,
                                      nullptr, nullptr, hidden, (int)N, 256, 1);
    wmma_gemm_k<<<gN, B, 0, stream>>>(hidden, ginW2 + (size_t)i * 128 * 128,
                                      nullptr, nullptr, h, (int)N, 128, 0);
    // BatchNorm (training stats, biased var)
    fill_k<<<1, 256, 0, stream>>>(stats, 256, 0.0f);
    bn_stats_k<<<1024, B, 0, stream>>>(h, stats, N * 128);
    bn_apply_k<<<blk(N * 128), B, 0, stream>>>(h, stats, bng + i * 128, bnb + i * 128,
                                               hloc, N * 128, 1.0f / (float)N);
    // GCN on sh
    wmma_gemm_k<<<gN, B, 0, stream>>>(sh, gcnW + (size_t)i * 128 * 128,
                                      nullptr, nullptr, sw, (int)N, 128, 0);
    gcn_base_k<<<blk(N * 128), B, 0, stream>>>(sw, dinv, gcnb + i * 128, shn, N * 128);
    gcn_scatter_k<<<blk(E * 32), B, 0, stream>>>(src, dst, sw, dinv, shn, E);
    tanh_k<<<blk(N * 128), B, 0, stream>>>(shn, sh, N * 128);
  }

  // head: y = concat(h_local, sh) @ W_hp + b_hp ; pool ; FF
  concat2_k<<<blk(N * 128), B, 0, stream>>>(hloc, sh, hc, nullptr, N * 128);
  wmma_gemm_k<<<gN, B, 0, stream>>>(hc, W_hp, b_hp, nullptr, y, (int)N, 256, 0);
  fill_k<<<blk(G * 128), B, 0, stream>>>(g, G * 128, 0.0f);
  pool_k<<<blk(N * 128), B, 0, stream>>>(y, batch, g, N * 128);

  wmma_gemm_k<<<gG, B, 0, stream>>>(g,   ffW1, nullptr, nullptr, gt1, (int)G, 128, 1);
  wmma_gemm_k<<<gG, B, 0, stream>>>(gt1, ffW2, nullptr, nullptr, gt2, (int)G, 128, 1);
  wmma_gemm_k<<<gG, B, 0, stream>>>(gt2, ffW3, nullptr, nullptr, gt1, (int)G, 128, 1);
  wmma_gemm_k<<<gG, B, 0, stream>>>(g,   ffWs, nullptr, gt1,     out, (int)G, 128, 0);
}